// Aligner_46136538694284
// MI455X (gfx1250) — compile-verified
//
#include <hip/hip_runtime.h>
#include <math.h>

// Problem constants (from the reference)
#define B_     16
#define T_TOK  512
#define T_AUD  4096
#define D_     256
#define INV_T2 0.01f   // TEMPERATURE^-2, 10^-2

typedef __attribute__((ext_vector_type(2))) float v2f;
typedef __attribute__((ext_vector_type(8))) float v8f;

// ---------------------------------------------------------------------------
// Kernel 1: W[b,t] = sum_a softmax_t( -0.01*(centers[b,t]-ts[b,a])^2 )
// Grid: (chunks, B). Block: 256 threads = 8 wave32 waves.
// Lane L owns tokens {i*32+L}, i=0..15 (bank-conflict-free LDS access).
// All accumulation in registers; cross-wave and cross-chunk partials reduced
// in a fixed order -> fully deterministic (no float atomics anywhere).
// ---------------------------------------------------------------------------
__global__ __launch_bounds__(256) void aligner_weights_kernel(
    const float* __restrict__ centers,   // [B][T_TOK]
    const float* __restrict__ audio_ts,  // [B][T_AUD]
    float* __restrict__ wpart,           // [chunks][B][T_TOK]
    int chunks)
{
  __shared__ float sC[T_TOK];
  __shared__ float sAcc[8][T_TOK];

  const int b     = blockIdx.y;
  const int chunk = blockIdx.x;
  const int tid   = threadIdx.x;
  const int wave  = tid >> 5;
  const int lane  = tid & 31;

  sC[tid]       = centers[b * T_TOK + tid];
  sC[tid + 256] = centers[b * T_TOK + tid + 256];
  __syncthreads();

  float acc[16];
#pragma unroll
  for (int i = 0; i < 16; ++i) acc[i] = 0.f;

  const int fpc = T_AUD / chunks;        // frames per chunk (chunks is 2^k)
  const int a0  = chunk * fpc;

  for (int a = a0 + wave; a < a0 + fpc; a += 8) {
    const float ta = audio_ts[b * T_AUD + a];
    float e[16];
    float m = -3.0e38f;
#pragma unroll
    for (int i = 0; i < 16; ++i) {
      const float dd = sC[i * 32 + lane] - ta;
      const float l  = -INV_T2 * dd * dd;
      e[i] = l;
      m    = fmaxf(m, l);
    }
    // wave32 max-reduce
#pragma unroll
    for (int off = 16; off >= 1; off >>= 1)
      m = fmaxf(m, __shfl_xor(m, off, 32));

    float s = 0.f;
#pragma unroll
    for (int i = 0; i < 16; ++i) { e[i] = __expf(e[i] - m); s += e[i]; }
    // wave32 sum-reduce
#pragma unroll
    for (int off = 16; off >= 1; off >>= 1)
      s += __shfl_xor(s, off, 32);

    const float inv = 1.0f / s;
#pragma unroll
    for (int i = 0; i < 16; ++i) acc[i] = fmaf(e[i], inv, acc[i]);
  }

#pragma unroll
  for (int i = 0; i < 16; ++i) sAcc[wave][i * 32 + lane] = acc[i];
  __syncthreads();

  // fixed-order reduction across the 8 waves
  for (int t = tid; t < T_TOK; t += 256) {
    float s = 0.f;
#pragma unroll
    for (int w = 0; w < 8; ++w) s += sAcc[w][t];
    wpart[(chunk * B_ + b) * T_TOK + t] = s;
  }
}

// ---------------------------------------------------------------------------
// Kernel 2: out[b, :] = H[b] (256x512) @ W[b] (512) via V_WMMA_F32_16X16X4_F32.
// Grid: B blocks of 512 threads = 16 waves; wave w computes d-tile rows
// [16w, 16w+16). B-matrix columns all hold w[k0..k0+3] (broadcast), so every
// output column equals the matvec result; lanes with N==0 store it.
// A 16x4 f32 fragment: lanes 0-15 -> M=lane, K={0,1}; lanes 16-31 -> K={2,3}.
// ---------------------------------------------------------------------------
__global__ __launch_bounds__(512) void aligner_out_kernel(
    const float* __restrict__ hidden,  // [B][D][T_TOK]
    const float* __restrict__ wpart,   // [chunks][B][T_TOK]
    float* __restrict__ out,           // [B][D]
    int chunks)
{
  __shared__ float sW[T_TOK];
  const int b   = blockIdx.x;
  const int tid = threadIdx.x;

  // deterministic fixed-order chunk reduction into LDS
  {
    float s = 0.f;
    for (int c = 0; c < chunks; ++c) s += wpart[(c * B_ + b) * T_TOK + tid];
    sW[tid] = s;
  }
  __syncthreads();

  const int wave = tid >> 5;       // d-tile index 0..15
  const int lane = tid & 31;
  const int mrow = lane & 15;
  const int koff = (lane < 16) ? 0 : 2;

  const float* __restrict__ hrow =
      hidden + (size_t)(b * D_ + wave * 16 + mrow) * T_TOK + koff;

  v8f c = {0.f, 0.f, 0.f, 0.f, 0.f, 0.f, 0.f, 0.f};
  for (int k0 = 0; k0 < T_TOK; k0 += 4) {
    v2f a;
    a.x = hrow[k0];
    a.y = hrow[k0 + 1];
    v2f bb;
    bb.x = sW[k0 + koff];
    bb.y = sW[k0 + koff + 1];
    // 8 args: (neg_a, A, neg_b, B, c_mod, C, reuse_a, reuse_b)
    c = __builtin_amdgcn_wmma_f32_16x16x4_f32(false, a, false, bb,
                                              (short)0, c, false, false);
  }

  // D layout: VGPR r = row M=r (lanes 0-15, N=lane) / M=r+8 (lanes 16-31).
  // N==0 lives in lane 0 (rows 0-7) and lane 16 (rows 8-15).
  if (lane == 0) {
#pragma unroll
    for (int r = 0; r < 8; ++r) out[b * D_ + wave * 16 + r] = c[r];
  } else if (lane == 16) {
#pragma unroll
    for (int r = 0; r < 8; ++r) out[b * D_ + wave * 16 + 8 + r] = c[r];
  }
}

// ---------------------------------------------------------------------------
extern "C" void kernel_launch(void* const* d_in, const int* in_sizes, int n_in,
                              void* d_out, int out_size, void* d_ws, size_t ws_size,
                              hipStream_t stream) {
  (void)in_sizes; (void)n_in; (void)out_size;

  const float* hidden  = (const float*)d_in[0];  // [16][256][512] f32
  const float* centers = (const float*)d_in[1];  // [16][512]      f32
  const float* ats     = (const float*)d_in[2];  // [16][4096]     f32
  float* out = (float*)d_out;                    // [16][256]      f32
  float* ws  = (float*)d_ws;                     // partial W buffers

  // Pick chunk count (power of 2, <=16) that fits in workspace; deterministic
  // given the same ws_size.
  const size_t bytes_per_chunk = (size_t)B_ * T_TOK * sizeof(float);
  int chunks = 16;
  while (chunks > 1 && ws_size < (size_t)chunks * bytes_per_chunk) chunks >>= 1;

  dim3 gridW(chunks, B_);
  aligner_weights_kernel<<<gridW, dim3(256), 0, stream>>>(centers, ats, ws, chunks);
  aligner_out_kernel<<<dim3(B_), dim3(512), 0, stream>>>(hidden, ws, out, chunks);
}